// WaveLetUnPooling_81561428951817
// MI455X (gfx1250) — compile-verified
//
#include <hip/hip_runtime.h>

typedef __attribute__((ext_vector_type(4))) float v4f;

// ---------------------------------------------------------------------------
// Kernel A: Haar 2x2 upsample of the four band inputs -> channels [0,512).
// Output-driven: thread t handles one float4 of the output.
//   t = pix*128 + qc ; qc = band*32 + c4 ; pix = ((b*256)+y)*256+x
// Each wave32 = 32 consecutive qc within one band of one pixel:
//   * one contiguous 512B global_load_b128 burst from the band input
//   * one contiguous 512B non-temporal global_store_b128 burst to the output
// Input re-reads (x/x+1, row 2i/2i+1) hit the 192MB L2.
// ---------------------------------------------------------------------------
__global__ __launch_bounds__(256) void haar_upwave_kernel(
    const v4f* __restrict__ LL, const v4f* __restrict__ LH,
    const v4f* __restrict__ HL, const v4f* __restrict__ HH,
    v4f* __restrict__ out)
{
    unsigned t    = blockIdx.x * 256u + threadIdx.x;   // < 33,554,432
    unsigned qc   = t & 127u;                          // float4 within 512 band chans
    unsigned pix  = t >> 7;                            // < 4*256*256
    unsigned band = qc >> 5;                           // 0=LL 1=LH 2=HL 3=HH
    unsigned c4   = qc & 31u;                          // float4 within band (C=128)

    unsigned b = pix >> 16;
    unsigned y = (pix >> 8) & 255u;
    unsigned x = pix & 255u;
    unsigned i = y >> 1, j = x >> 1;

    const v4f* src = (band == 0) ? LL : (band == 1) ? LH : (band == 2) ? HL : HH;
    unsigned sidx = ((b * 128u + i) * 128u + j) * 32u + c4;
    v4f v = src[sidx];                                 // cached (RT) load: 4x L2 reuse

    // k[p][q] = 0.5 * rowF[y&1] * colF[x&1], H = [-1,+1]
    float s = 0.5f;
    if ((band & 1u) && !(x & 1u)) s = -s;              // LH/HH: column high-pass
    if ((band & 2u) && !(y & 1u)) s = -s;              // HL/HH: row high-pass
    v = v * s;

    __builtin_nontemporal_store(v, out + (pix * 160u + qc));  // 671MB stream: NT
}

// ---------------------------------------------------------------------------
// Kernel B: passthrough tensor_in -> channels [512,640).
// Pure 134MB data movement: use the CDNA5 async global<->LDS DMA path so the
// copy runs on the async engines (ASYNCcnt), 512B per wave per instruction,
// 4 transfers in flight per wave.
//   q (float4 idx) : src byte = q*16 ; pix = q>>5, c4 = q&31
//   dst byte       = pix*2560 + 2048 + c4*16   (640ch * 4B, +512ch offset)
// ---------------------------------------------------------------------------
#define COPY_UNROLL 4

__global__ __launch_bounds__(256) void tensor_copy_async_kernel(
    const float* __restrict__ src, float* __restrict__ dstBase)
{
    __shared__ __align__(16) float lbuf[256 * COPY_UNROLL * 4];   // 16KB staging
    // Generic shared-aperture address: low 32 bits == LDS byte address (ISA 10.2)
    unsigned lds_base = (unsigned)(unsigned long long)(&lbuf[0]);
    unsigned tid = threadIdx.x;

    unsigned q[COPY_UNROLL];
    unsigned laddr[COPY_UNROLL];
#pragma unroll
    for (int k = 0; k < COPY_UNROLL; ++k) {
        q[k]     = blockIdx.x * (256u * COPY_UNROLL) + (unsigned)k * 256u + tid;
        laddr[k] = lds_base + ((unsigned)k * 256u + tid) * 16u;
        unsigned soff = q[k] * 16u;                     // < 2^31
        asm volatile("global_load_async_to_lds_b128 %0, %1, %2"
                     :: "v"(laddr[k]), "v"(soff), "s"(src)
                     : "memory");
    }
    // All async loads complete (LDS written) before stores read it back.
    asm volatile("s_wait_asynccnt 0" ::: "memory");
#pragma unroll
    for (int k = 0; k < COPY_UNROLL; ++k) {
        unsigned pix  = q[k] >> 5;
        unsigned c4   = q[k] & 31u;
        unsigned doff = pix * 2560u + 2048u + c4 * 16u; // < 2^31
        asm volatile("global_store_async_from_lds_b128 %0, %1, %2"
                     :: "v"(doff), "v"(laddr[k]), "s"(dstBase)
                     : "memory");
    }
    asm volatile("s_wait_asynccnt 0" ::: "memory");
}

// ---------------------------------------------------------------------------
// Launch: inputs in setup_inputs() order: LL_in, LH_in, HL_in, HH_in, tensor_in
// Output: [4,256,256,640] fp32 = 167,772,160 floats.
// ---------------------------------------------------------------------------
extern "C" void kernel_launch(void* const* d_in, const int* in_sizes, int n_in,
                              void* d_out, int out_size, void* d_ws, size_t ws_size,
                              hipStream_t stream) {
    (void)in_sizes; (void)n_in; (void)out_size; (void)d_ws; (void)ws_size;

    const v4f*   LL  = (const v4f*)d_in[0];
    const v4f*   LH  = (const v4f*)d_in[1];
    const v4f*   HL  = (const v4f*)d_in[2];
    const v4f*   HH  = (const v4f*)d_in[3];
    const float* tin = (const float*)d_in[4];
    float*       out = (float*)d_out;

    // Kernel A: 4*256*256*128 float4 outputs = 33,554,432 threads
    haar_upwave_kernel<<<131072, 256, 0, stream>>>(LL, LH, HL, HH, (v4f*)out);

    // Kernel B: 4*256*256*32 = 8,388,608 float4 copies, 4 per thread
    tensor_copy_async_kernel<<<8192, 256, 0, stream>>>(tin, out);
}